// SeDGPL1_43026982371461
// MI455X (gfx1250) — compile-verified
//
#include <hip/hip_runtime.h>
#include <hip/hip_bf16.h>
#include <math.h>

#define BB 16
#define SS 512
#define EE 32
#define HH 768
#define VV 50265

typedef __attribute__((ext_vector_type(2))) float v2f;
typedef __attribute__((ext_vector_type(8))) float v8f;

// ---------------------------------------------------------------------------
// Phase 1: gated fusion for the e==0 slice only (the only slice that matters).
// One block per batch. o2 (16 x 768, row-major) written to workspace.
// ---------------------------------------------------------------------------
__global__ __launch_bounds__(256) void sedgpl_gate_kernel(
    const int* __restrict__ batch_arg,     // (B,S)
    const int* __restrict__ mask_indices,  // (B,)  == event_pos[:,0]
    const float* __restrict__ sent_emb,    // (B,E,H)
    const float* __restrict__ type_emb,    // (B,S,H)
    const float* __restrict__ emb_table,   // (V,H)
    const float* __restrict__ W1_1,        // (H,H)
    const float* __restrict__ W1_2,
    const float* __restrict__ W2_1,
    const float* __restrict__ W2_2,
    float* __restrict__ o2_out)            // (B,H)
{
    __shared__ float s_inst[HH];
    __shared__ float s_sent[HH];
    __shared__ float s_typ[HH];
    __shared__ float s_o1[HH];

    const int b = blockIdx.x;
    const int t = threadIdx.x;

    const int pos = mask_indices[b];
    const int tok = batch_arg[b * SS + pos];

    const float* inst = emb_table + (size_t)tok * HH;
    const float* sent = sent_emb + (size_t)b * EE * HH;            // e == 0
    const float* typ  = type_emb + ((size_t)b * SS + pos) * HH;

    for (int h = t; h < HH; h += 256) {
        s_inst[h] = inst[h];
        s_sent[h] = sent[h];
        s_typ[h]  = typ[h];
    }
    __syncthreads();

    // gate 1: g1 = sigmoid(inst@W1_1 + sent@W1_2); o1 = g1*inst + (1-g1)*sent
    float o1v[3];
#pragma unroll
    for (int i = 0; i < 3; ++i) {
        const int h = t + i * 256;
        float acc = 0.0f;
        for (int k = 0; k < HH; ++k)
            acc = fmaf(s_inst[k], W1_1[(size_t)k * HH + h],
                  fmaf(s_sent[k], W1_2[(size_t)k * HH + h], acc));
        const float g = 1.0f / (1.0f + expf(-acc));
        o1v[i] = g * s_inst[h] + (1.0f - g) * s_sent[h];
    }
    __syncthreads();
#pragma unroll
    for (int i = 0; i < 3; ++i) s_o1[t + i * 256] = o1v[i];
    __syncthreads();

    // gate 2: g2 = sigmoid(o1@W2_1 + typ@W2_2); o2 = g2*o1 + (1-g2)*typ
#pragma unroll
    for (int i = 0; i < 3; ++i) {
        const int h = t + i * 256;
        float acc = 0.0f;
        for (int k = 0; k < HH; ++k)
            acc = fmaf(s_o1[k], W2_1[(size_t)k * HH + h],
                  fmaf(s_typ[k], W2_2[(size_t)k * HH + h], acc));
        const float g = 1.0f / (1.0f + expf(-acc));
        o2_out[(size_t)b * HH + h] = g * s_o1[h] + (1.0f - g) * s_typ[h];
    }
}

// ---------------------------------------------------------------------------
// Phase 2: prediction = o2(16x768) @ lm_W(768x50265) + lm_b, fp32 WMMA.
// HBM-bound on lm_W (~154 MB); one 16x16x4 f32 WMMA per K-step per wave.
// Block = 8 waves = 128 N-columns; 393 blocks cover V=50265.
// ---------------------------------------------------------------------------
__global__ __launch_bounds__(256) void sedgpl_lmhead_wmma_kernel(
    const float* __restrict__ o2,      // (16,768) row-major
    const float* __restrict__ lm_W,    // (768,V) row-major
    const float* __restrict__ lm_b,    // (V,)
    float* __restrict__ out)           // (16,V)
{
    const int lane = threadIdx.x & 31;
    const int wave = threadIdx.x >> 5;

    const int nl = lane & 15;                 // column within the 16-wide tile
    const int n  = blockIdx.x * 128 + wave * 16 + nl;
    const int nc = (n < VV) ? n : (VV - 1);   // clamped address: EXEC must stay all-1s

    // f32 WMMA operand layout:
    //  A 16x4:  lanes 0-15 -> M=lane, K offsets {0,1}; lanes 16-31 -> K offsets {2,3}
    //  B 4x16:  lanes 0-15 -> N=lane, K offsets {0,1}; lanes 16-31 -> K offsets {2,3}
    const int klo = (lane < 16) ? 0 : 2;

    const float* aptr = o2 + (size_t)nl * HH + klo;   // M = lane%16, contiguous K pair
    const float* bptr = lm_W + (size_t)klo * VV + nc;

    v8f c = {0.f, 0.f, 0.f, 0.f, 0.f, 0.f, 0.f, 0.f};

#pragma unroll 4
    for (int k0 = 0; k0 < HH; k0 += 4) {
        v2f a = *(const v2f*)(aptr + k0);             // one b64 load
        v2f bv;
        bv.x = bptr[(size_t)k0 * VV];                 // row K = k0 + klo
        bv.y = bptr[(size_t)(k0 + 1) * VV];           // row K = k0 + klo + 1
        // (neg_a, A, neg_b, B, c_mod, C, reuse_a, reuse_b)
        c = __builtin_amdgcn_wmma_f32_16x16x4_f32(
                false, a, false, bv, (short)0, c, false, false);
    }

    // D layout: lanes 0-15 -> M = 0..7 in VGPR 0..7; lanes 16-31 -> M = 8..15
    if (n < VV) {
        const float bias = lm_b[n];
        const int mbase = (lane < 16) ? 0 : 8;
#pragma unroll
        for (int i = 0; i < 8; ++i)
            out[(size_t)(mbase + i) * VV + n] = c[i] + bias;
    }
}

// ---------------------------------------------------------------------------
// Launch
// ---------------------------------------------------------------------------
extern "C" void kernel_launch(void* const* d_in, const int* in_sizes, int n_in,
                              void* d_out, int out_size, void* d_ws, size_t ws_size,
                              hipStream_t stream) {
    const int*   batch_arg    = (const int*)  d_in[0];
    // d_in[1] = event_pos : unused (mask_indices == event_pos[:,0])
    const int*   mask_indices = (const int*)  d_in[2];
    const float* sent_emb     = (const float*)d_in[3];
    const float* type_emb     = (const float*)d_in[4];
    const float* emb_table    = (const float*)d_in[5];
    const float* W1_1         = (const float*)d_in[6];
    const float* W1_2         = (const float*)d_in[7];
    const float* W2_1         = (const float*)d_in[8];
    const float* W2_2         = (const float*)d_in[9];
    const float* lm_W         = (const float*)d_in[10];
    const float* lm_b         = (const float*)d_in[11];
    float* out = (float*)d_out;

    float* o2_ws = (float*)d_ws;  // 16*768 floats = 48 KB

    sedgpl_gate_kernel<<<BB, 256, 0, stream>>>(
        batch_arg, mask_indices, sent_emb, type_emb, emb_table,
        W1_1, W1_2, W2_1, W2_2, o2_ws);

    const int nblocks = (VV + 127) / 128;  // 393
    sedgpl_lmhead_wmma_kernel<<<nblocks, 256, 0, stream>>>(
        o2_ws, lm_W, lm_b, out);
}